// KDTreeSampleLayer_70677981823085
// MI455X (gfx1250) — compile-verified
//
#include <hip/hip_runtime.h>

typedef __attribute__((ext_vector_type(2))) float v2f;
typedef __attribute__((ext_vector_type(4))) float v4f;
typedef __attribute__((ext_vector_type(8))) float v8f;

#define NB 8
#define NPTS 16384
#define NQ 2048
#define KNN 16

// ---------------------------------------------------------------------------
// Kernel 1: Farthest Point Sampling.
// One block per batch. 512 threads; each thread owns 32 points in registers
// (coords + running min sq-dist). Per iteration: distance update vs. last
// picked point, then wave-shfl argmax + LDS cross-wave argmax (tie -> lowest
// index, matching jnp.argmax). Writes pts (b, q, 3) directly.
// ---------------------------------------------------------------------------
#define FPS_THREADS 512
#define FPS_PPT (NPTS / FPS_THREADS) // 32 points per thread

__global__ __launch_bounds__(FPS_THREADS) void fps_kernel(
    const float* __restrict__ xyz, float* __restrict__ pts_out) {
  const int b = blockIdx.x;
  const int tid = threadIdx.x;
  const int lane = tid & 31;
  const int wave = tid >> 5;

  __shared__ float s_wd[32];
  __shared__ int s_wi[32];
  __shared__ int s_cur;

  const float* base = xyz + (size_t)b * NPTS * 3;
  float* pout = pts_out + (size_t)b * NQ * 3;

  float px[FPS_PPT], py[FPS_PPT], pz[FPS_PPT], mind[FPS_PPT];
#pragma unroll
  for (int j = 0; j < FPS_PPT; ++j) {
    int n = tid * FPS_PPT + j;
    px[j] = base[n * 3 + 0];
    py[j] = base[n * 3 + 1];
    pz[j] = base[n * 3 + 2];
    mind[j] = 1e10f;
  }

  if (tid == 0) {
    // First emitted sample is index 0 (reference scan emits carry first).
    pout[0] = base[0];
    pout[1] = base[1];
    pout[2] = base[2];
    s_cur = 0;
  }
  if (wave == 0) { s_wd[lane] = -1e30f; s_wi[lane] = 0x7fffffff; }
  __syncthreads();
  int cur = 0;

  for (int i = 1; i < NQ; ++i) {
    const float lx = base[cur * 3 + 0];
    const float ly = base[cur * 3 + 1];
    const float lz = base[cur * 3 + 2];

    float bd = -1.0f;
    int bi = 0;
#pragma unroll
    for (int j = 0; j < FPS_PPT; ++j) {
      float dx = px[j] - lx, dy = py[j] - ly, dz = pz[j] - lz;
      float d = dx * dx + dy * dy + dz * dz;
      float m = fminf(mind[j], d);
      mind[j] = m;
      if (m > bd) { bd = m; bi = tid * FPS_PPT + j; }
    }
    // wave32 argmax reduce (max dist, tie -> min index)
#pragma unroll
    for (int off = 16; off >= 1; off >>= 1) {
      float od = __shfl_xor(bd, off, 32);
      int oi = __shfl_xor(bi, off, 32);
      if (od > bd || (od == bd && oi < bi)) { bd = od; bi = oi; }
    }
    if (lane == 0) { s_wd[wave] = bd; s_wi[wave] = bi; }
    __syncthreads();
    if (wave == 0) {
      bd = s_wd[lane];
      bi = s_wi[lane];
#pragma unroll
      for (int off = 16; off >= 1; off >>= 1) {
        float od = __shfl_xor(bd, off, 32);
        int oi = __shfl_xor(bi, off, 32);
        if (od > bd || (od == bd && oi < bi)) { bd = od; bi = oi; }
      }
      if (lane == 0) {
        s_cur = bi;
        pout[i * 3 + 0] = base[bi * 3 + 0];
        pout[i * 3 + 1] = base[bi * 3 + 1];
        pout[i * 3 + 2] = base[bi * 3 + 2];
      }
    }
    __syncthreads();
    cur = s_cur;
  }
}

// ---------------------------------------------------------------------------
// Kernel 2: 16-NN via V_WMMA_F32_16X16X4_F32.
// Ranking metric: ||x||^2 - 2 p.x  (== d2 - ||p||^2; per-query constant shift
// preserves top-k order), computed as a single 16x16x4 WMMA:
//   A[m,:] = (-2px, -2py, -2pz, 1)    B[:,n] = (x, y, z, ||x||^2)
// Each wave owns 32 queries (two A tiles). Per round: two B column tiles
// (32 points, global loads software-pipelined one round ahead), 4 WMMAs,
// dump D column-major to LDS with a 36-float column stride (36*lm mod 64
// covers banks {0,4,...,60} -> the 16 b128 stores tile all 64 banks
// conflict-free, and 144B stride keeps 16B alignment), then each lane
// batch-loads its 32 candidates (DS loads clause under one wait) and merges
// them into a register-resident sorted top-16.
// ---------------------------------------------------------------------------
#define KNN_WAVES 4
#define KNN_THREADS (KNN_WAVES * 32)
#define QPB (KNN_WAVES * 32) // 128 queries per block
#define CSTRIDE 36           // LDS column stride in floats (bank-conflict-free)

__global__ __launch_bounds__(KNN_THREADS) void knn_kernel(
    const float* __restrict__ xyz, const float* __restrict__ pts,
    int* __restrict__ idx_out) {
  const int b = blockIdx.x >> 4;            // 16 blocks per batch
  const int qblk = (blockIdx.x & 15) * QPB; // query block base
  const int tid = threadIdx.x;
  const int lane = tid & 31;
  const int wave = tid >> 5;
  const int qbase = qblk + wave * 32;

  __shared__ __align__(16) float s_d[KNN_WAVES][32 * CSTRIDE];
  float* sd = &s_d[wave][0];

  const float* xb = xyz + (size_t)b * NPTS * 3;
  const float* pb = pts + (size_t)b * NQ * 3;

  const int half = lane >> 4; // 0: lanes 0-15 (K=0,1) ; 1: lanes 16-31 (K=2,3)
  const int lm = lane & 15;

  // A matrices (16x4 f32: VGPR0 holds K=0|2, VGPR1 holds K=1|3 per lane half)
  v2f a0, a1;
  {
    const float* q0 = pb + (size_t)(qbase + lm) * 3;
    const float* q1 = pb + (size_t)(qbase + 16 + lm) * 3;
    if (half == 0) {
      a0.x = -2.0f * q0[0]; a0.y = -2.0f * q0[1];
      a1.x = -2.0f * q1[0]; a1.y = -2.0f * q1[1];
    } else {
      a0.x = -2.0f * q0[2]; a0.y = 1.0f;
      a1.x = -2.0f * q1[2]; a1.y = 1.0f;
    }
  }

  // B tile loader (4x16 f32, mirrored layout: VGPR0=K0|K2, VGPR1=K1|K3)
  auto loadB = [&](int nc, v2f& b0, v2f& b1) {
    const float* p0 = xb + (size_t)(nc + lm) * 3;
    const float* p1 = xb + (size_t)(nc + 16 + lm) * 3;
    if (half == 0) {
      b0.x = p0[0]; b0.y = p0[1];
      b1.x = p1[0]; b1.y = p1[1];
    } else {
      float x0 = p0[0], y0 = p0[1], z0 = p0[2];
      float x1 = p1[0], y1 = p1[1], z1 = p1[2];
      b0.x = z0; b0.y = x0 * x0 + y0 * y0 + z0 * z0;
      b1.x = z1; b1.y = x1 * x1 + y1 * y1 + z1 * z1;
    }
  };

  float topd[KNN];
  int topi[KNN];
#pragma unroll
  for (int j = 0; j < KNN; ++j) { topd[j] = 1e30f; topi[j] = 0; }

  v2f bv0, bv1;
  loadB(0, bv0, bv1);

  for (int nc = 0; nc < NPTS; nc += 32) {
    // Prefetch next round's point tiles while this round computes.
    v2f nb0 = bv0, nb1 = bv1;
    if (nc + 32 < NPTS) loadB(nc + 32, nb0, nb1);

    v8f c00 = {}, c01 = {}, c10 = {}, c11 = {};
    c00 = __builtin_amdgcn_wmma_f32_16x16x4_f32(false, a0, false, bv0,
                                                (short)0, c00, false, false);
    c01 = __builtin_amdgcn_wmma_f32_16x16x4_f32(false, a0, false, bv1,
                                                (short)0, c01, false, false);
    c10 = __builtin_amdgcn_wmma_f32_16x16x4_f32(false, a1, false, bv0,
                                                (short)0, c10, false, false);
    c11 = __builtin_amdgcn_wmma_f32_16x16x4_f32(false, a1, false, bv1,
                                                (short)0, c11, false, false);

    // D layout: VGPR v -> row M=v (lanes 0-15) or M=v+8 (lanes 16-31), col=lm.
    // Store column-major: sd[col*CSTRIDE + rowInWave], rowInWave = 16*tile + M.
    {
      float* d00 = sd + lm * CSTRIDE + 8 * half;        // tile0, col nc+lm
      float* d10 = d00 + 16;                            // tile1, col nc+lm
      float* d01 = sd + (16 + lm) * CSTRIDE + 8 * half; // tile0, col nc+16+lm
      float* d11 = d01 + 16;                            // tile1, col nc+16+lm
      *(v4f*)(d00) = v4f{c00[0], c00[1], c00[2], c00[3]};
      *(v4f*)(d00 + 4) = v4f{c00[4], c00[5], c00[6], c00[7]};
      *(v4f*)(d10) = v4f{c10[0], c10[1], c10[2], c10[3]};
      *(v4f*)(d10 + 4) = v4f{c10[4], c10[5], c10[6], c10[7]};
      *(v4f*)(d01) = v4f{c01[0], c01[1], c01[2], c01[3]};
      *(v4f*)(d01 + 4) = v4f{c01[4], c01[5], c01[6], c01[7]};
      *(v4f*)(d11) = v4f{c11[0], c11[1], c11[2], c11[3]};
      *(v4f*)(d11 + 4) = v4f{c11[4], c11[5], c11[6], c11[7]};
    }
    // Wave-synchronous LDS reuse: wait for DS stores, block compiler reorder.
    asm volatile("s_wait_dscnt 0" ::: "memory");

    // Batch-load this lane's 32 candidates (row `lane`) so the DS loads can
    // issue as a clause under a single dscnt wait, then merge from registers.
    float cand[32];
#pragma unroll
    for (int c = 0; c < 32; ++c) cand[c] = sd[c * CSTRIDE + lane];

#pragma unroll
    for (int c = 0; c < 32; ++c) {
      float d = cand[c];
      int n = nc + c;
      if (d < topd[KNN - 1]) {
        float dj = d;
        int ij = n;
#pragma unroll
        for (int j = 0; j < KNN; ++j) {
          bool sw = dj < topd[j];
          float td = topd[j];
          int ti = topi[j];
          topd[j] = sw ? dj : td;
          topi[j] = sw ? ij : ti;
          dj = sw ? td : dj;
          ij = sw ? ti : ij;
        }
      }
    }
    asm volatile("" ::: "memory"); // reads complete before next round's stores

    bv0 = nb0;
    bv1 = nb1;
  }

  // Emit sorted (ascending d2) neighbor indices: (b, q, 16)
  const int q = qbase + lane;
  int* op = idx_out + ((size_t)b * NQ + q) * KNN;
#pragma unroll
  for (int j = 0; j < KNN; ++j) op[j] = topi[j];
}

extern "C" void kernel_launch(void* const* d_in, const int* in_sizes, int n_in,
                              void* d_out, int out_size, void* d_ws,
                              size_t ws_size, hipStream_t stream) {
  (void)in_sizes; (void)n_in; (void)d_ws; (void)ws_size; (void)out_size;
  const float* xyz = (const float*)d_in[0];
  // d_in[1] = pointCount (16), d_in[2] = nqueries (2048): compile-time here.

  int* idx_out = (int*)d_out;                             // (8,2048,16) i32
  float* pts_out = (float*)d_out + (size_t)NB * NQ * KNN; // (8,2048,3) f32

  fps_kernel<<<NB, FPS_THREADS, 0, stream>>>(xyz, pts_out);
  knn_kernel<<<NB * (NQ / QPB), KNN_THREADS, 0, stream>>>(xyz, pts_out,
                                                          idx_out);
}